// NodeModel_82652350644752
// MI455X (gfx1250) — compile-verified
//
#include <hip/hip_runtime.h>
#include <math.h>

// ---------------------------------------------------------------------------
// GraphNet edge/node block on MI455X (gfx1250, wave32).
// Matrix math via V_WMMA_F32_16X16X4_F32 (fp32, K=4 exact fit for the 4->20
// edge layer). Random gather (x: 12MB) and scatter accumulators (16MB) are
// L2-resident (192MB); streamed inputs use non-temporal hints.
// Waves are fully independent: per-wave LDS slices, wave-local fences only
// (DS ops are in-order per wave) -> scatter atomics pipeline across tiles
// with no storecnt drains and no s_barrier traffic.
// ---------------------------------------------------------------------------

typedef __attribute__((ext_vector_type(2))) float v2f;
typedef __attribute__((ext_vector_type(8))) float v8f;

#define WAVES_PER_BLOCK 4
#define TILES_PER_WAVE  8

__device__ __forceinline__ v8f wmma4(v2f a, v2f b, v8f c) {
    return __builtin_amdgcn_wmma_f32_16x16x4_f32(
        false, a, false, b, (short)0, c, false, false);
}

// Wave-local LDS exchange fence: compiler ordering only. Hardware DS ops are
// processed in issue order within a wave, so no s_barrier / counter drain is
// required for intra-wave cross-lane exchange through LDS.
__device__ __forceinline__ void wave_fence() {
    __builtin_amdgcn_fence(__ATOMIC_ACQ_REL, "wavefront");
    __builtin_amdgcn_wave_barrier();
}

// ---------------------------------------------------------------------------
__global__ void zero_f32v4(float4* __restrict__ p, int n4) {
    int i = blockIdx.x * blockDim.x + threadIdx.x;
    if (i < n4) p[i] = make_float4(0.f, 0.f, 0.f, 0.f);
}

// ---------------------------------------------------------------------------
// Edge kernel: 4 independent waves per block, TILES_PER_WAVE x 16 edges/wave.
//   h   = [x[row], edge_attr]    (16x4)
//   hid = relu(h @ w1a + b1a)    (16x20) -> 2 WMMA (N tiles 0-15 / 16-19)
//   o   = hid @ w1b + b1b        (16x3)  -> 5 chained K=4 WMMA
//   atomicAdd into s[col], cnt[col]
// ---------------------------------------------------------------------------
__global__ void edge_kernel(const float* __restrict__ x,
                            const int*   __restrict__ ei,     // [2,E]
                            const float* __restrict__ eattr,  // [E]
                            const float* __restrict__ w1a,    // [4,20]
                            const float* __restrict__ b1a,    // [20]
                            const float* __restrict__ w1b,    // [20,3]
                            const float* __restrict__ b1b,    // [3]
                            float* __restrict__ s,            // [N,3]
                            float* __restrict__ cnt,          // [N]
                            int E) {
    __shared__ float hA_s[WAVES_PER_BLOCK][16 * 4];
    __shared__ int   scol_s[WAVES_PER_BLOCK][16];
    __shared__ float hH_s[WAVES_PER_BLOCK][16 * 36];  // padded stride, bank-safe

    const int wid  = threadIdx.x >> 5;
    const int lane = threadIdx.x & 31;
    const int m    = lane & 15;      // row (M) / col (N) index
    const int half = lane >> 4;
    const int kb   = half * 2;       // K base of A/B fragments

    float* hA   = hA_s[wid];
    int*   scol = scol_s[wid];
    float* hH   = hH_s[wid];

    // ---- hoisted weight fragments (loaded once per wave) ----
    v2f B10 = { w1a[kb * 20 + m], w1a[(kb + 1) * 20 + m] };    // hid cols 0..15
    const int  n2  = 16 + m;                                   // hid cols 16..19
    const bool n2v = (n2 < 20);
    const int  n2c = n2v ? n2 : 19;
    v2f B11 = { n2v ? w1a[kb * 20 + n2c] : 0.0f,
                n2v ? w1a[(kb + 1) * 20 + n2c] : 0.0f };
    const float bia  = b1a[m];
    const float bia2 = n2v ? b1a[n2c] : 0.0f;

    const bool mv = (m < 3);
    const int  mc = mv ? m : 2;
    v2f B2f[5];
#pragma unroll
    for (int kk = 0; kk < 5; ++kk) {
        int K0 = kk * 4 + kb;
        float u0 = w1b[K0 * 3 + mc];
        float u1 = w1b[(K0 + 1) * 3 + mc];
        B2f[kk].x = mv ? u0 : 0.0f;
        B2f[kk].y = mv ? u1 : 0.0f;
    }
    const float bb = b1b[mc];

    const int tile0 = (blockIdx.x * WAVES_PER_BLOCK + wid) * TILES_PER_WAVE;
    for (int t = 0; t < TILES_PER_WAVE; ++t) {
        const int ebase = (tile0 + t) * 16;
        const int e     = ebase + m;
        const bool ev   = (e < E);
        const int  ec   = ev ? e : 0;

        // ---- gather: both lane halves load in parallel ----
        int row = __builtin_nontemporal_load(&ei[ec]);
        if (half == 0) {
            hA[m * 4 + 0] = ev ? x[row * 3 + 0] : 0.0f;
            hA[m * 4 + 1] = ev ? x[row * 3 + 1] : 0.0f;
            int col = __builtin_nontemporal_load(&ei[E + ec]);
            scol[m] = ev ? col : -1;
            if (ev) atomicAdd(&cnt[col], 1.0f);
        } else {
            hA[m * 4 + 2] = ev ? x[row * 3 + 2] : 0.0f;
            hA[m * 4 + 3] = ev ? __builtin_nontemporal_load(&eattr[ec]) : 0.0f;
        }
        wave_fence();

        // ---- layer 1: (16x4)@(4x20), K=4 exact ----
        v2f A1 = { hA[m * 4 + kb], hA[m * 4 + kb + 1] };
        v8f z  = {};
        v8f d0 = wmma4(A1, B10, z);
        v8f d1 = wmma4(A1, B11, z);

        // bias+relu, relayout D(MxN) -> row-major LDS (unconditional stores)
#pragma unroll
        for (int k = 0; k < 8; ++k) {
            int r = k + half * 8;             // C/D: VGPR k holds M = k+8*half
            hH[r * 36 + m]  = fmaxf(d0[k] + bia, 0.0f);
            hH[r * 36 + n2] = n2v ? fmaxf(d1[k] + bia2, 0.0f) : 0.0f;
        }
        wave_fence();

        // ---- layer 2: (16x20)@(20x3), five K=4 chunks ----
        v8f c = z;
#pragma unroll
        for (int kk = 0; kk < 5; ++kk) {
            int K0 = kk * 4 + kb;
            v2f A2 = { hH[m * 36 + K0], hH[m * 36 + K0 + 1] };
            c = wmma4(A2, B2f[kk], c);
        }

        // ---- scatter-add (output cols 0..2); atomics pipeline across tiles
        if (mv) {
#pragma unroll
            for (int k = 0; k < 8; ++k) {
                int r   = k + half * 8;
                int col = scol[r];
                if (col >= 0) atomicAdd(&s[col * 3 + m], c[k] + bb);
            }
        }
        wave_fence();
    }
}

// ---------------------------------------------------------------------------
// Node kernel: 4 independent waves per block, TILES_PER_WAVE x 16 nodes/wave.
//   agg = s/max(cnt,1); h = [x, agg] (16x6, K padded to 8)
//   hid = relu(h @ w2a + b2a) -> 4 WMMA ; o = hid @ w2b + b2b -> 5 WMMA
//   out = o / ||o||_2
// ---------------------------------------------------------------------------
__global__ void node_kernel(const float* __restrict__ x,
                            const float* __restrict__ s,
                            const float* __restrict__ cnt,
                            const float* __restrict__ w2a,   // [6,20]
                            const float* __restrict__ b2a,   // [20]
                            const float* __restrict__ w2b,   // [20,3]
                            const float* __restrict__ b2b,   // [3]
                            float* __restrict__ out,         // [N,3]
                            int N) {
    __shared__ float hA_s[WAVES_PER_BLOCK][16 * 9];   // padded, bank-safe
    __shared__ float hH_s[WAVES_PER_BLOCK][16 * 36];
    __shared__ float oL_s[WAVES_PER_BLOCK][16 * 3];

    const int wid  = threadIdx.x >> 5;
    const int lane = threadIdx.x & 31;
    const int m    = lane & 15;
    const int half = lane >> 4;
    const int kb   = half * 2;

    float* hA = hA_s[wid];
    float* hH = hH_s[wid];
    float* oL = oL_s[wid];

    // ---- hoisted weight fragments ----
    const int  n2  = 16 + m;
    const bool n2v = (n2 < 20);
    const int  n2c = n2v ? n2 : 19;

    v2f Bf0[2], Bf1[2];              // layer-1 B, two K-chunks x two N-tiles
#pragma unroll
    for (int kk = 0; kk < 2; ++kk) {
        int K0 = kk * 4 + kb;        // 0..7; valid K < 6
        int K0c = (K0 < 6) ? K0 : 5;
        int K1c = (K0 + 1 < 6) ? K0 + 1 : 5;
        float a0 = w2a[K0c * 20 + m];
        float a1 = w2a[K1c * 20 + m];
        float b0 = w2a[K0c * 20 + n2c];
        float b1 = w2a[K1c * 20 + n2c];
        Bf0[kk].x = (K0 < 6)     ? a0 : 0.0f;
        Bf0[kk].y = (K0 + 1 < 6) ? a1 : 0.0f;
        Bf1[kk].x = (K0 < 6 && n2v)     ? b0 : 0.0f;
        Bf1[kk].y = (K0 + 1 < 6 && n2v) ? b1 : 0.0f;
    }
    const float bia  = b2a[m];
    const float bia2 = n2v ? b2a[n2c] : 0.0f;

    const bool mv = (m < 3);
    const int  mc = mv ? m : 2;
    v2f B2f[5];
#pragma unroll
    for (int kk = 0; kk < 5; ++kk) {
        int K0 = kk * 4 + kb;
        float u0 = w2b[K0 * 3 + mc];
        float u1 = w2b[(K0 + 1) * 3 + mc];
        B2f[kk].x = mv ? u0 : 0.0f;
        B2f[kk].y = mv ? u1 : 0.0f;
    }
    const float bb = b2b[mc];

    const int tile0 = (blockIdx.x * WAVES_PER_BLOCK + wid) * TILES_PER_WAVE;
    for (int t = 0; t < TILES_PER_WAVE; ++t) {
        const int base = (tile0 + t) * 16;
        const int i    = base + m;
        const bool iv  = (i < N);
        const int  ic  = iv ? i : 0;

        // ---- load stage: both halves in parallel ----
        float cdiv = fmaxf(cnt[ic], 1.0f);
        if (half == 0) {
            hA[m * 9 + 0] = iv ? x[ic * 3 + 0] : 0.0f;
            hA[m * 9 + 1] = iv ? x[ic * 3 + 1] : 0.0f;
            hA[m * 9 + 2] = iv ? x[ic * 3 + 2] : 0.0f;
            hA[m * 9 + 6] = 0.0f;
            hA[m * 9 + 7] = 0.0f;
        } else {
            hA[m * 9 + 3] = iv ? s[ic * 3 + 0] / cdiv : 0.0f;
            hA[m * 9 + 4] = iv ? s[ic * 3 + 1] / cdiv : 0.0f;
            hA[m * 9 + 5] = iv ? s[ic * 3 + 2] / cdiv : 0.0f;
        }
        wave_fence();

        // ---- layer 1: (16x6->8)@(6x20) ----
        v8f z = {};
        v8f d0 = z, d1 = z;
#pragma unroll
        for (int kk = 0; kk < 2; ++kk) {
            int K0 = kk * 4 + kb;
            v2f A = { hA[m * 9 + K0], hA[m * 9 + K0 + 1] };
            d0 = wmma4(A, Bf0[kk], d0);
            d1 = wmma4(A, Bf1[kk], d1);
        }
#pragma unroll
        for (int k = 0; k < 8; ++k) {
            int r = k + half * 8;
            hH[r * 36 + m]  = fmaxf(d0[k] + bia, 0.0f);
            hH[r * 36 + n2] = n2v ? fmaxf(d1[k] + bia2, 0.0f) : 0.0f;
        }
        wave_fence();

        // ---- layer 2: (16x20)@(20x3) ----
        v8f c = z;
#pragma unroll
        for (int kk = 0; kk < 5; ++kk) {
            int K0 = kk * 4 + kb;
            v2f A = { hH[m * 36 + K0], hH[m * 36 + K0 + 1] };
            c = wmma4(A, B2f[kk], c);
        }

        if (mv) {
#pragma unroll
            for (int k = 0; k < 8; ++k) {
                int r = k + half * 8;
                oL[r * 3 + m] = c[k] + bb;
            }
        }
        wave_fence();

        // ---- row-wise L2 normalize + NT store ----
        if (half == 0 && iv) {
            float o0 = oL[m * 3 + 0];
            float o1 = oL[m * 3 + 1];
            float o2 = oL[m * 3 + 2];
            float fac = sqrtf(o0 * o0 + o1 * o1 + o2 * o2);
            __builtin_nontemporal_store(o0 / fac, &out[i * 3 + 0]);
            __builtin_nontemporal_store(o1 / fac, &out[i * 3 + 1]);
            __builtin_nontemporal_store(o2 / fac, &out[i * 3 + 2]);
        }
        wave_fence();
    }
}

// ---------------------------------------------------------------------------
extern "C" void kernel_launch(void* const* d_in, const int* in_sizes, int n_in,
                              void* d_out, int out_size, void* d_ws, size_t ws_size,
                              hipStream_t stream) {
    // setup_inputs() order:
    // 0:x 1:edge_index 2:edge_attr 3:u 4:batch
    // 5:w1a 6:b1a 7:w1b 8:b1b 9:w2a 10:b2a 11:w2b 12:b2b
    const float* x     = (const float*)d_in[0];
    const int*   ei    = (const int*)  d_in[1];
    const float* eattr = (const float*)d_in[2];
    const float* w1a   = (const float*)d_in[5];
    const float* b1a   = (const float*)d_in[6];
    const float* w1b   = (const float*)d_in[7];
    const float* b1b   = (const float*)d_in[8];
    const float* w2a   = (const float*)d_in[9];
    const float* b2a   = (const float*)d_in[10];
    const float* w2b   = (const float*)d_in[11];
    const float* b2b   = (const float*)d_in[12];

    const int N = in_sizes[0] / 3;
    const int E = in_sizes[2];

    float* s   = (float*)d_ws;            // [N,3] scatter accumulator
    float* cnt = s + (size_t)3 * N;       // [N]   edge counts
    float* out = (float*)d_out;

    // zero accumulators (4N floats) with B128 stores
    {
        int n4 = (4 * N + 3) / 4;
        zero_f32v4<<<(n4 + 255) / 256, 256, 0, stream>>>((float4*)s, n4);
    }

    // edge pass
    {
        int ntiles  = (E + 15) / 16;
        int nwaves  = (ntiles + TILES_PER_WAVE - 1) / TILES_PER_WAVE;
        int nblocks = (nwaves + WAVES_PER_BLOCK - 1) / WAVES_PER_BLOCK;
        edge_kernel<<<nblocks, 32 * WAVES_PER_BLOCK, 0, stream>>>(
            x, ei, eattr, w1a, b1a, w1b, b1b, s, cnt, E);
    }

    // node pass
    {
        int ntiles  = (N + 15) / 16;
        int nwaves  = (ntiles + TILES_PER_WAVE - 1) / TILES_PER_WAVE;
        int nblocks = (nwaves + WAVES_PER_BLOCK - 1) / WAVES_PER_BLOCK;
        node_kernel<<<nblocks, 32 * WAVES_PER_BLOCK, 0, stream>>>(
            x, s, cnt, w2a, b2a, w2b, b2b, out, N);
    }
}